// AdditiveRecursiveNN_46351287058652
// MI455X (gfx1250) — compile-verified
//
#include <hip/hip_runtime.h>
#include <hip/hip_bf16.h>
#include <math.h>

// ---------------------------------------------------------------------------
// AdditiveRecursiveNN on MI455X (gfx1250, wave32).
//
// Tree of 64x64 fp32 matmuls. Memory-bound (~16 FLOP/byte vs 23.3 TB/s HBM),
// so we keep fp32 fidelity and use V_WMMA_F32_16X16X4_F32 for the matmuls.
// One 256-thread block (8 waves) per tree node:
//   - TDM (tensor_load_to_lds) DMAs the 16 KB W tile (embedding gather) into
//     LDS while the waves compute s = h_l + h_r into LDS (32 KB total)
//   - each wave computes two 16x16 D-tiles: 16 K-steps of wmma_f32_16x16x4_f32
//   - epilogue fuses per-column bias + ReLU
// ---------------------------------------------------------------------------

#define E        64
#define EE       4096      // E*E
#define N_LEAF   4096
#define TPB      256

typedef float v2f __attribute__((ext_vector_type(2)));
typedef float v8f __attribute__((ext_vector_type(8)));
typedef unsigned int v4u __attribute__((ext_vector_type(4)));
typedef int v4i __attribute__((ext_vector_type(4)));
typedef int v8i __attribute__((ext_vector_type(8)));

#if defined(__has_builtin)
#if __has_builtin(__builtin_amdgcn_tensor_load_to_lds) && \
    __has_builtin(__builtin_amdgcn_s_wait_tensorcnt)
#define USE_TDM 1
#endif
#endif
#ifndef USE_TDM
#define USE_TDM 0
#endif

// ---------------------------------------------------------------------------
// Leaf init: h[i] = relu(embedding[node_ids[N_LEAF-1 + i]])   (4096 blocks)
// ---------------------------------------------------------------------------
__global__ void tree_leaf_kernel(const int* __restrict__ node_ids,
                                 const float* __restrict__ embedding,
                                 float* __restrict__ h_out) {
    const int leaf = blockIdx.x;
    const int t    = threadIdx.x;
    const size_t id = (size_t)node_ids[(N_LEAF - 1) + leaf];
    const float* __restrict__ src = embedding + id * EE;
    float* __restrict__ dst = h_out + (size_t)leaf * EE;
#pragma unroll
    for (int r = 0; r < 4; ++r) {
        const int i4 = (r * TPB + t) * 4;
        float4 v = *(const float4*)(src + i4);
        v.x = fmaxf(v.x, 0.0f);
        v.y = fmaxf(v.y, 0.0f);
        v.z = fmaxf(v.z, 0.0f);
        v.w = fmaxf(v.w, 0.0f);
        *(float4*)(dst + i4) = v;
    }
}

// ---------------------------------------------------------------------------
// One tree level: for node b in [0, n):
//   s = h_in[2b] + h_in[2b+1]
//   h_out[b] = relu(W(node_ids[start+b]) @ s + bias[:,None broadcast over rows])
// Grid = n blocks, 256 threads each.
// ---------------------------------------------------------------------------
__global__ void tree_level_kernel(const int* __restrict__ node_ids,
                                  int start,
                                  const float* __restrict__ embedding,
                                  const float* __restrict__ bias_table,
                                  const float* __restrict__ h_in,
                                  float* __restrict__ h_out) {
    __shared__ float sW[EE];   // 16 KB: W, row-major 64x64
    __shared__ float sS[EE];   // 16 KB: s = h_left + h_right, row-major 64x64

    const int node = blockIdx.x;
    const int t    = threadIdx.x;
    const size_t id = (size_t)node_ids[start + node];

    const float* __restrict__ W  = embedding + id * EE;
    const float* __restrict__ h0 = h_in + (size_t)(2 * node) * EE;
    const float* __restrict__ h1 = h0 + EE;

#if USE_TDM
    // ------------------------------------------------------------------
    // TDM: one DMA op moves the whole 64x64 fp32 W tile into LDS.
    // D# per CDNA5 ISA ch.8: group0 = {count/flags, lds_addr, global_addr,
    // type}, group1 = {data_size, dims, tile dims, strides}. 2D tensor, so
    // groups 2/3 are zero. Issued from wave 0 only (TDM ignores EXEC).
    // ------------------------------------------------------------------
    if (t < 32) {
        const unsigned long long gaddr = (unsigned long long)(const void*)W;
        const unsigned lds_addr = (unsigned)(size_t)(const void*)sW; // low 32b = LDS byte addr

        v4u g0;
        g0.x = 1u;                                           // count=1, is_restore=0, gather off
        g0.y = lds_addr;                                     // lds_addr[31:0]
        g0.z = (unsigned)(gaddr & 0xFFFFFFFFull);            // global_addr[31:0]
        g0.w = (unsigned)((gaddr >> 32) & 0x1FFFFFFull)      // global_addr[56:32]
               | 0x80000000u;                                // type=2 ("image")

        v8i g1;
        g1[0] = 0x00020000;          // workgroup_mask=0, data_size=2 (4 bytes)
        g1[1] = (E << 16);           // tensor_dim0[15:0]=64 at bits 63:48
        g1[2] = (E << 16);           // tensor_dim0 hi=0, tensor_dim1[15:0]=64
        g1[3] = (E << 16);           // tensor_dim1 hi=0, tile_dim0=64
        g1[4] = E;                   // tile_dim1=64, tile_dim2=0 (2D)
        g1[5] = E;                   // tensor_dim0_stride[31:0]=64 elements
        g1[6] = 0;                   // stride0 hi=0, tensor_dim1_stride lo=0 (2D)
        g1[7] = 0;

        v4i gz4 = {0, 0, 0, 0};
        v8i gz8 = {0, 0, 0, 0, 0, 0, 0, 0};
        // 6-arg toolchain form: (g0, g1, g2, g3, g4, cpol)
        __builtin_amdgcn_tensor_load_to_lds(g0, g1, gz4, gz4, gz8, 0);
    }
#else
    // Fallback staging of W through VGPRs.
#pragma unroll
    for (int r = 0; r < 4; ++r) {
        const int i4 = (r * TPB + t) * 4;
        *(float4*)(sW + i4) = *(const float4*)(W + i4);
    }
#endif

    // Overlapped with the TDM: stage s = h_left + h_right into LDS.
#pragma unroll
    for (int r = 0; r < 4; ++r) {
        const int i4 = (r * TPB + t) * 4;
        float4 a = *(const float4*)(h0 + i4);
        float4 b = *(const float4*)(h1 + i4);
        float4 s;
        s.x = a.x + b.x; s.y = a.y + b.y; s.z = a.z + b.z; s.w = a.w + b.w;
        *(float4*)(sS + i4) = s;
    }

#if USE_TDM
    __builtin_amdgcn_s_wait_tensorcnt(0);   // W tile landed in LDS (wave 0's DMA)
#endif
    __syncthreads();

    const int wave = t >> 5;        // 0..7
    const int lane = t & 31;
    const int lh   = lane & 15;     // lane within half-wave
    const int hi   = lane >> 4;     // 0 = lanes 0-15, 1 = lanes 16-31

    float* __restrict__ outBase = h_out + (size_t)node * EE;

    // 16 output tiles (4x4 grid of 16x16); each wave owns tiles {wave, wave+8}.
#pragma unroll
    for (int tt = 0; tt < 2; ++tt) {
        const int tile = wave + tt * 8;
        const int ti = tile >> 2;   // tile row  (M block)
        const int tj = tile & 3;    // tile col  (N block)

        v8f acc = {};               // C = 0, accumulate over K

#pragma unroll
        for (int kk = 0; kk < 16; ++kk) {
            // ISA layout, 32-bit A 16x4: lanes 0-15 hold K=0,1 ; lanes 16-31 hold K=2,3
            const int k4 = kk * 4 + hi * 2;
            v2f a;
            a.x = sW[(ti * 16 + lh) * E + k4 + 0];
            a.y = sW[(ti * 16 + lh) * E + k4 + 1];
            // B 4x16: rows striped across lanes within a VGPR, same K split per half
            v2f b;
            b.x = sS[(k4 + 0) * E + tj * 16 + lh];
            b.y = sS[(k4 + 1) * E + tj * 16 + lh];
            // (neg_a, A, neg_b, B, c_mod, C, reuse_a, reuse_b)
            acc = __builtin_amdgcn_wmma_f32_16x16x4_f32(
                    false, a, false, b, (short)0, acc, false, false);
        }

        // Epilogue: bias is per-column (b[:, None, :] in the reference), then ReLU.
        const float bias = bias_table[id * E + (size_t)(tj * 16 + lh)];
#pragma unroll
        for (int r = 0; r < 8; ++r) {
            float v = acc[r] + bias;
            v = fmaxf(v, 0.0f);
            // D 16x16 f32 layout: lane<16 -> rows 0-7, lane>=16 -> rows 8-15
            const int row = ti * 16 + hi * 8 + r;
            const int col = tj * 16 + lh;
            outBase[row * E + col] = v;
        }
    }
}

// ---------------------------------------------------------------------------
// Head: logits = root @ proj_w.T + proj_b ; loss = -log_softmax(logits)[label]
// prediction = argmax(logits).  One block.
// ---------------------------------------------------------------------------
__global__ void tree_head_kernel(const float* __restrict__ root,
                                 const float* __restrict__ proj_w,
                                 const float* __restrict__ proj_b,
                                 const int* __restrict__ label,
                                 float* __restrict__ out) {
    __shared__ float red0[TPB];
    __shared__ float red1[TPB];
    const int t = threadIdx.x;
    float p0 = 0.0f, p1 = 0.0f;
#pragma unroll
    for (int i = t; i < EE; i += TPB) {
        const float r = root[i];
        p0 += r * proj_w[i];        // row 0 of (L, E*E)
        p1 += r * proj_w[EE + i];   // row 1
    }
    red0[t] = p0;
    red1[t] = p1;
    __syncthreads();
    for (int s = TPB / 2; s > 0; s >>= 1) {
        if (t < s) {
            red0[t] += red0[t + s];
            red1[t] += red1[t + s];
        }
        __syncthreads();
    }
    if (t == 0) {
        const float l0 = red0[0] + proj_b[0];
        const float l1 = red1[0] + proj_b[1];
        const int pred = (l1 > l0) ? 1 : 0;   // argmax, first index wins ties
        const float m   = fmaxf(l0, l1);
        const float lse = m + logf(expf(l0 - m) + expf(l1 - m));
        const int   lb  = label[0];
        const float loss = lse - ((lb == 0) ? l0 : l1);
        out[0] = (float)pred;
        out[1] = loss;
    }
}

// ---------------------------------------------------------------------------
// Host launcher. Inputs (setup_inputs order):
//   0 node_ids (int32, 8191)   1 label (int32, 1)
//   2 embedding (f32, V*4096)  3 bias_table (f32, V*64)
//   4 proj_w (f32, 2*4096)     5 proj_b (f32, 2)
// Workspace: ping-pong h buffers, buf0 = 64 MiB (4096 nodes), buf1 = 32 MiB.
// ---------------------------------------------------------------------------
extern "C" void kernel_launch(void* const* d_in, const int* in_sizes, int n_in,
                              void* d_out, int out_size, void* d_ws, size_t ws_size,
                              hipStream_t stream) {
    const int*   node_ids   = (const int*)d_in[0];
    const int*   label      = (const int*)d_in[1];
    const float* embedding  = (const float*)d_in[2];
    const float* bias_table = (const float*)d_in[3];
    const float* proj_w     = (const float*)d_in[4];
    const float* proj_b     = (const float*)d_in[5];
    float*       out        = (float*)d_out;

    float* buf0 = (float*)d_ws;                                       // 4096 * 4096 f32
    float* buf1 = (float*)((char*)d_ws + (size_t)N_LEAF * EE * 4);    // 2048 * 4096 f32

    // Leaves: h = relu(embedding[leaf_ids])
    tree_leaf_kernel<<<N_LEAF, TPB, 0, stream>>>(node_ids, embedding, buf0);

    // Levels D-2 .. 0  (11 .. 0): ping-pong buf0 <-> buf1
    float* cur = buf0;
    float* nxt = buf1;
    for (int level = 11; level >= 0; --level) {
        const int n     = 1 << level;
        const int start = n - 1;
        tree_level_kernel<<<n, TPB, 0, stream>>>(node_ids, start, embedding,
                                                 bias_table, cur, nxt);
        float* tmp = cur; cur = nxt; nxt = tmp;
    }

    // Head: logits / loss / argmax from root matrix (in `cur`)
    tree_head_kernel<<<1, TPB, 0, stream>>>(cur, proj_w, proj_b, label, out);
}